// LightGCN_5265629905633
// MI455X (gfx1250) — compile-verified
//
#include <hip/hip_runtime.h>
#include <hip/hip_bf16.h>

typedef __attribute__((ext_vector_type(2))) float v2f;
typedef __attribute__((ext_vector_type(8))) float v8f;

#define DMODEL 256
#define LDS_STRIDE 260   // 256 + 4-float pad: A-fragment reads hit 16 distinct LDS banks

// ---------------------------------------------------------------------------
// h = X * W  via V_WMMA_F32_16X16X4_F32.
// One block = one 16-row M tile staged in LDS. 8 waves; each wave owns a
// 16x32 output strip (two 16x16 N-tiles computed concurrently so every
// A fragment fetched from LDS feeds TWO WMMAs).
// A 16x4 f32 fragment (ISA layout): lane l holds M=l&15, K-pair = (l>>4)*2.
// B 4x16 f32 fragment (mirrored):   lane l holds N=l&15, K-pair = (l>>4)*2.
// C/D 16x16 f32: vgpr r of lane l = element (M = r + 8*(l>>4), N = l&15).
// ---------------------------------------------------------------------------
__global__ __launch_bounds__(256) void gcn_gemm_wmma(const float* __restrict__ x,
                                                     const float* __restrict__ W,
                                                     float* __restrict__ h,
                                                     int nrows) {
    __shared__ float xs[16 * LDS_STRIDE];

    const int tid = threadIdx.x;
    const int m0  = blockIdx.x * 16;

    // Cooperative load of the 16x256 X tile (float4 granularity: 1024 vec4s).
    const float4* __restrict__ x4 = (const float4*)x;
    float4* xs4 = (float4*)xs;               // LDS_STRIDE is a multiple of 4
    for (int i = tid; i < 16 * (DMODEL / 4); i += 256) {
        const int r  = i >> 6;               // row 0..15
        const int c4 = i & 63;               // float4 column
        float4 v = make_float4(0.f, 0.f, 0.f, 0.f);
        const int gr = m0 + r;
        if (gr < nrows) v = x4[(size_t)gr * (DMODEL / 4) + c4];
        xs4[r * (LDS_STRIDE / 4) + c4] = v;
    }
    __syncthreads();

    const int lane  = tid & 31;
    const int mrow  = lane & 15;             // A: M index / B,C: N index
    const int kb    = (lane >> 4) << 1;      // K sub-pair: 0 or 2
    const int n0    = (tid >> 5) * 32;       // wave's 32-column strip
    const int ncol0 = n0 + mrow;
    const int ncol1 = n0 + 16 + mrow;

    v8f acc0 = {};
    v8f acc1 = {};
    #pragma unroll 4
    for (int k = 0; k < DMODEL; k += 4) {
        // One shared A fragment (8B-aligned pair -> ds_load_b64, conflict-free)
        v2f a = *(const v2f*)(&xs[mrow * LDS_STRIDE + k + kb]);
        // Two B fragments from global W (256 KB, L2-hot)
        const float* __restrict__ wr0 = W + (size_t)(k + kb) * DMODEL;
        const float* __restrict__ wr1 = wr0 + DMODEL;
        v2f b0, b1;
        b0.x = wr0[ncol0];  b0.y = wr1[ncol0];
        b1.x = wr0[ncol1];  b1.y = wr1[ncol1];
        acc0 = __builtin_amdgcn_wmma_f32_16x16x4_f32(
            false, a, false, b0, (short)0, acc0, false, false);
        acc1 = __builtin_amdgcn_wmma_f32_16x16x4_f32(
            false, a, false, b1, (short)0, acc1, false, false);
    }

    // Store D tiles: contiguous 64B per half-wave per row.
    const int mbase = m0 + ((lane >> 4) << 3);
    if (m0 + 16 <= nrows) {                  // uniform fast path (N % 16 == 0)
        #pragma unroll
        for (int r = 0; r < 8; ++r) {
            float* __restrict__ hr = h + (size_t)(mbase + r) * DMODEL;
            hr[ncol0] = acc0[r];
            hr[ncol1] = acc1[r];
        }
    } else {                                 // generic guarded tail
        #pragma unroll
        for (int r = 0; r < 8; ++r) {
            if (mbase + r < nrows) {
                float* __restrict__ hr = h + (size_t)(mbase + r) * DMODEL;
                hr[ncol0] = acc0[r];
                hr[ncol1] = acc1[r];
            }
        }
    }
}

// ---------------------------------------------------------------------------
// Utility: zero a float buffer (d_out / deg are poisoned, must be cleared).
// ---------------------------------------------------------------------------
__global__ __launch_bounds__(256) void gcn_zero_f32(float* __restrict__ p, int n) {
    const int i = blockIdx.x * 256 + threadIdx.x;
    if (i < n) p[i] = 0.0f;
}

// Degree of target nodes: deg[dst[e]] += 1.
__global__ __launch_bounds__(256) void gcn_degree(const int* __restrict__ dst,
                                                  float* __restrict__ deg, int E) {
    const int e = blockIdx.x * 256 + threadIdx.x;
    if (e < E) atomicAdd(&deg[dst[e]], 1.0f);
}

// In place: deg -> deg>0 ? rsqrt(deg) : 0  (deg>=1 when >0, so max(deg,1)=deg).
__global__ __launch_bounds__(256) void gcn_inv_sqrt(float* __restrict__ deg, int n) {
    const int i = blockIdx.x * 256 + threadIdx.x;
    if (i < n) {
        const float d = deg[i];
        deg[i] = (d > 0.0f) ? rsqrtf(d) : 0.0f;
    }
}

// ---------------------------------------------------------------------------
// Scatter: one wave per edge. Wave-uniform src/dst/norm; each lane handles
// 8 floats, float4 columns lane and lane+32 (2x float4 loads of h[src], 8 f32 atomic adds to out[dst]).
// h and out are L2-resident (51 MB each vs 192 MB L2).
// ---------------------------------------------------------------------------
__global__ __launch_bounds__(256) void gcn_scatter(const int* __restrict__ src,
                                                   const int* __restrict__ dst,
                                                   const float* __restrict__ h,
                                                   const float* __restrict__ dis,
                                                   float* __restrict__ out, int E) {
    const int wid  = (blockIdx.x * 256 + threadIdx.x) >> 5;   // edge id
    const int lane = threadIdx.x & 31;
    if (wid >= E) return;

    const int s = src[wid];
    const int d = dst[wid];
    const float norm = dis[s] * dis[d];

    const float4* __restrict__ hs = (const float4*)(h + (size_t)s * DMODEL);
    float* od = out + (size_t)d * DMODEL;

    #pragma unroll
    for (int j = 0; j < 2; ++j) {
        const int c4 = lane + 32 * j;              // float4 column 0..63
        const float4 v = hs[c4];
        const int base = c4 * 4;
        atomicAdd(od + base + 0, v.x * norm);
        atomicAdd(od + base + 1, v.y * norm);
        atomicAdd(od + base + 2, v.z * norm);
        atomicAdd(od + base + 3, v.w * norm);
    }
}

// ---------------------------------------------------------------------------
extern "C" void kernel_launch(void* const* d_in, const int* in_sizes, int n_in,
                              void* d_out, int out_size, void* d_ws, size_t ws_size,
                              hipStream_t stream) {
    const float* x  = (const float*)d_in[0];
    const float* W  = (const float*)d_in[1];
    const int*   ei = (const int*)d_in[2];    // edge_index [2, E], row-major

    const int N = in_sizes[0] / DMODEL;       // 50000
    const int E = in_sizes[2] / 2;            // 1600000
    const int* src = ei;
    const int* dst = ei + E;

    float* out = (float*)d_out;
    float* h   = (float*)d_ws;                           // [N, 256] fp32
    float* deg = h + (size_t)N * DMODEL;                 // [N] fp32 (-> deg_inv_sqrt)

    // 1) h = X * W (f32 WMMA)
    gcn_gemm_wmma<<<(N + 15) / 16, 256, 0, stream>>>(x, W, h, N);

    // 2) degrees of targets
    gcn_zero_f32<<<(N + 255) / 256, 256, 0, stream>>>(deg, N);
    gcn_degree<<<(E + 255) / 256, 256, 0, stream>>>(dst, deg, E);
    gcn_inv_sqrt<<<(N + 255) / 256, 256, 0, stream>>>(deg, N);

    // 3) out = 0, then scatter-add normalized messages (one wave per edge)
    const int outN = N * DMODEL;
    gcn_zero_f32<<<(outN + 255) / 256, 256, 0, stream>>>(out, outN);
    const long long threads = (long long)E * 32;
    gcn_scatter<<<(int)((threads + 255) / 256), 256, 0, stream>>>(src, dst, h, deg, out, E);
}